// MultiHeadSelfAttention_48275432407843
// MI455X (gfx1250) — compile-verified
//
#include <hip/hip_runtime.h>
#include <hip/hip_bf16.h>

typedef __attribute__((ext_vector_type(2))) float v2f;
typedef __attribute__((ext_vector_type(8))) float v8f;

#define EMBED 768
#define S_LEN 2048
#define NHEAD 12
#define HDIM  64
#define BATCH 2
#define ROWS  (BATCH * S_LEN)   /* 4096 */

__device__ __forceinline__ v8f wmma4(v2f a, v2f b, v8f c) {
  // D = A(16x4,f32) * B(4x16,f32) + C(16x16,f32)
  return __builtin_amdgcn_wmma_f32_16x16x4_f32(
      /*neg_a=*/false, a, /*neg_b=*/false, b,
      /*c_mod=*/(short)0, c, /*reuse_a=*/false, /*reuse_b=*/false);
}

__device__ __forceinline__ v8f vzero8() {
  v8f z = {0.f, 0.f, 0.f, 0.f, 0.f, 0.f, 0.f, 0.f};
  return z;
}

// ---------------------------------------------------------------------------
// C[M,N] = A[M,K] * W[K,N] (+ bias), M=4096, N=768, K=768, all row-major fp32.
// Block = 256 threads (8 waves). Wave tile = 16(M) x 64(N) (4 accumulators).
// Block tile = 64(M) x 128(N). Grid = (4096/64, 768/128) = (64, 6).
// ---------------------------------------------------------------------------
__global__ __launch_bounds__(256) void gemm_bias_kernel(
    const float* __restrict__ A, const float* __restrict__ W,
    const float* __restrict__ bias, float* __restrict__ C, int has_bias) {
  const int lane = threadIdx.x & 31;
  const int wave = threadIdx.x >> 5;
  const int half = lane >> 4;   // K-half selector for A/B fragments
  const int l16  = lane & 15;

  const int m0 = blockIdx.x * 64  + (wave & 3) * 16;
  const int n0 = blockIdx.y * 128 + (wave >> 2) * 64;

  v8f acc[4];
#pragma unroll
  for (int i = 0; i < 4; ++i) acc[i] = vzero8();

  const float* arow = A + (size_t)(m0 + l16) * EMBED;

#pragma unroll 2
  for (int kk = 0; kk < EMBED; kk += 4) {
    const int ka = kk + 2 * half;                  // K index for vgpr0 of frag
    const v2f a = *(const v2f*)(arow + ka);        // A[m=l16][ka], [ka+1]
    const float* w0 = W + (size_t)ka * EMBED + n0 + l16;  // B[k=ka][n]
    const float* w1 = w0 + EMBED;                          // B[k=ka+1][n]
#pragma unroll
    for (int nt = 0; nt < 4; ++nt) {
      v2f b;
      b.x = w0[nt * 16];
      b.y = w1[nt * 16];
      acc[nt] = wmma4(a, b, acc[nt]);
    }
  }

  // D layout: row = v + 8*half, col = n0 + nt*16 + l16
#pragma unroll
  for (int nt = 0; nt < 4; ++nt) {
    const int col = n0 + nt * 16 + l16;
    const float bv = has_bias ? bias[col] : 0.f;
#pragma unroll
    for (int v = 0; v < 8; ++v) {
      C[(size_t)(m0 + v + 8 * half) * EMBED + col] = acc[nt][v] + bv;
    }
  }
}

// ---------------------------------------------------------------------------
// Flash attention per head. Q,K,V laid out as [B, S, E] with this head's
// slice at column h*64. One wave owns one 16-row query tile; loops over 128
// key tiles with online softmax. P (16x16) is restaged via per-wave LDS to
// convert the WMMA C/D layout into the A-fragment layout for P*V.
// ---------------------------------------------------------------------------
__global__ __launch_bounds__(256) void attention_kernel(
    const float* __restrict__ Q, const float* __restrict__ K,
    const float* __restrict__ V, const float* __restrict__ mask,
    float* __restrict__ O) {
  __shared__ float pstage[8][16 * 16];   // 1KB per wave

  const int lane = threadIdx.x & 31;
  const int wave = threadIdx.x >> 5;
  const int half = lane >> 4;
  const int l16  = lane & 15;

  const int tile = blockIdx.x * 8 + wave;   // 0 .. 3071
  const int qt = tile & 127;                // q-tile within (b,h)
  const int bh = tile >> 7;                 // 0 .. 23
  const int b  = bh / NHEAD;
  const int h  = bh % NHEAD;

  const int    q0   = qt * 16;
  const size_t base = (size_t)b * S_LEN * EMBED + (size_t)h * HDIM;

  const float* qrow = Q + base + (size_t)(q0 + l16) * EMBED;

  v8f acc[4];
#pragma unroll
  for (int i = 0; i < 4; ++i) acc[i] = vzero8();
  float mrow[8], lrow[8];
#pragma unroll
  for (int v = 0; v < 8; ++v) { mrow[v] = -3.0e38f; lrow[v] = 0.f; }

  float* myp = &pstage[wave][0];

  for (int kt = 0; kt < S_LEN / 16; ++kt) {
    // ---- S(16x16) = Q_tile * K_tile^T ; B=K^T -> contiguous f2 from K rows
    v8f s = vzero8();
    const float* krow = K + base + (size_t)(kt * 16 + l16) * EMBED;
#pragma unroll
    for (int kk = 0; kk < HDIM; kk += 4) {
      v2f a  = *(const v2f*)(qrow + kk + 2 * half);
      v2f bf = *(const v2f*)(krow + kk + 2 * half);
      s = wmma4(a, bf, s);
    }

    // ---- scale + additive mask, online softmax (rows live in vgpr index)
    const float* mp = mask + (size_t)(q0 + 8 * half) * S_LEN + kt * 16 + l16;
    float p[8];
#pragma unroll
    for (int v = 0; v < 8; ++v) {
      float sv = s[v] * 0.125f + mp[(size_t)v * S_LEN] * (-1.0e9f);
      float rmax = sv;   // reduce over the 16 keys (lanes of this half)
      rmax = fmaxf(rmax, __shfl_xor(rmax, 1, 32));
      rmax = fmaxf(rmax, __shfl_xor(rmax, 2, 32));
      rmax = fmaxf(rmax, __shfl_xor(rmax, 4, 32));
      rmax = fmaxf(rmax, __shfl_xor(rmax, 8, 32));
      const float mnew  = fmaxf(mrow[v], rmax);
      const float alpha = __expf(mrow[v] - mnew);
      const float pv    = __expf(sv - mnew);
      float rsum = pv;
      rsum += __shfl_xor(rsum, 1, 32);
      rsum += __shfl_xor(rsum, 2, 32);
      rsum += __shfl_xor(rsum, 4, 32);
      rsum += __shfl_xor(rsum, 8, 32);
      lrow[v] = lrow[v] * alpha + rsum;
      mrow[v] = mnew;
      p[v] = pv;
#pragma unroll
      for (int nt = 0; nt < 4; ++nt) acc[nt][v] *= alpha;
    }

    // ---- stage P: P[row = v+8*half][key = l16]
#pragma unroll
    for (int v = 0; v < 8; ++v) myp[(v + 8 * half) * 16 + l16] = p[v];
    __syncthreads();

    // ---- O(16x64) += P(16x16) * V_tile(16x64)
#pragma unroll
    for (int kk = 0; kk < 16; kk += 4) {
      v2f a = *(const v2f*)(myp + l16 * 16 + kk + 2 * half);  // A = P
      const float* v0 = V + base + (size_t)(kt * 16 + kk + 2 * half) * EMBED + l16;
      const float* v1 = v0 + EMBED;
#pragma unroll
      for (int nt = 0; nt < 4; ++nt) {
        v2f bf;
        bf.x = v0[nt * 16];
        bf.y = v1[nt * 16];
        acc[nt] = wmma4(a, bf, acc[nt]);
      }
    }
    __syncthreads();
  }

  // ---- normalize by row sums, write concat layout O[b, q, h*64 + f]
#pragma unroll
  for (int v = 0; v < 8; ++v) {
    const float inv = 1.0f / lrow[v];
    float* orow = O + base + (size_t)(q0 + v + 8 * half) * EMBED;
#pragma unroll
    for (int nt = 0; nt < 4; ++nt) {
      orow[nt * 16 + l16] = acc[nt][v] * inv;
    }
  }
}

// ---------------------------------------------------------------------------
extern "C" void kernel_launch(void* const* d_in, const int* in_sizes, int n_in,
                              void* d_out, int out_size, void* d_ws, size_t ws_size,
                              hipStream_t stream) {
  const float* x    = (const float*)d_in[0];
  const float* mask = (const float*)d_in[1];
  const float* Wq   = (const float*)d_in[2];
  const float* Wk   = (const float*)d_in[3];
  const float* Wv   = (const float*)d_in[4];
  const float* Wo   = (const float*)d_in[5];
  const float* bo   = (const float*)d_in[6];

  float* ws = (float*)d_ws;
  const size_t mat = (size_t)ROWS * EMBED;   // 4096*768 floats
  float* Qb = ws;
  float* Kb = ws + mat;
  float* Vb = ws + 2 * mat;
  float* Ab = ws + 3 * mat;

  dim3 g(ROWS / 64, EMBED / 128);   // (64, 6)
  gemm_bias_kernel<<<g, 256, 0, stream>>>(x, Wq, nullptr, Qb, 0);
  gemm_bias_kernel<<<g, 256, 0, stream>>>(x, Wk, nullptr, Kb, 0);
  gemm_bias_kernel<<<g, 256, 0, stream>>>(x, Wv, nullptr, Vb, 0);

  const int qtiles = BATCH * NHEAD * (S_LEN / 16);   // 3072 waves
  attention_kernel<<<qtiles / 8, 256, 0, stream>>>(Qb, Kb, Vb, mask, Ab);

  gemm_bias_kernel<<<g, 256, 0, stream>>>(Ab, Wo, bo, (float*)d_out, 1);
}